// SFTLayer_21320217657612
// MI455X (gfx1250) — compile-verified
//
#include <hip/hip_runtime.h>

typedef __attribute__((ext_vector_type(2))) float v2f;
typedef __attribute__((ext_vector_type(4))) float v4f;
typedef __attribute__((ext_vector_type(8))) float v8f;

#define LEAKY_ALPHA 0.3f

// ---------------------------------------------------------------------------
// 3x3 SAME conv (NHWC, C=32->32) + bias + LeakyReLU, as implicit GEMM on
// V_WMMA_F32_16X16X4_F32.  One wave32 per 16-pixel (W-dim) x 32-channel tile.
// blockIdx.y picks the branch (gamma weights vs beta weights).
//   A[16x4]  : 16 pixels x 4 in-channels   (2 VGPRs/lane)
//   B[4x16]  : 4 in-channels x 16 out-ch   (2 VGPRs/lane)
//   C/D[16x16]: v8f accumulator, two N-halves for 32 out channels.
// SAME padding handled branchlessly: clamp halo addresses into bounds and
// multiply the loaded A values by a per-lane 0/1 mask, so no EXEC divergence
// and all loads stay unconditional (fusable into b64).
// ---------------------------------------------------------------------------
__global__ __launch_bounds__(32) void conv3x3_lrelu_wmma(
    const float* __restrict__ inA, const float* __restrict__ inB,
    const float* __restrict__ wA,  const float* __restrict__ biasA,
    const float* __restrict__ wB,  const float* __restrict__ biasB,
    float* __restrict__ outA, float* __restrict__ outB)
{
    constexpr int H = 64, W = 64, C = 32;

    const int lane = threadIdx.x;     // 0..31
    const int hlf  = lane >> 4;       // which 16-lane half
    const int l16  = lane & 15;       // A-row (pixel m) / B-col (out-ch n)

    const int tile = blockIdx.x;      // b*H*(W/16) + h*(W/16) + wt
    const int wt = tile & 3;
    const int h  = (tile >> 2) & (H - 1);
    const int b  = tile >> 8;
    const int w0 = wt * 16;

    const int br = blockIdx.y;
    const float* in   = br ? inB   : inA;
    const float* wgt  = br ? wB    : wA;
    const float* bias = br ? biasB : biasA;
    float*       out  = br ? outB  : outA;

    v8f acc0 = {};   // out channels 0..15
    v8f acc1 = {};   // out channels 16..31

    #pragma unroll
    for (int dy = -1; dy <= 1; ++dy) {
        const int hy  = h + dy;
        const int hyc = hy < 0 ? 0 : (hy > H - 1 ? H - 1 : hy);
        const bool hok = (hy >= 0) & (hy < H);
        #pragma unroll
        for (int dx = -1; dx <= 1; ++dx) {
            const int wx  = w0 + l16 + dx;             // pixel for A-row m=l16
            const int wxc = wx < 0 ? 0 : (wx > W - 1 ? W - 1 : wx);
            // 1.0 inside the image, 0.0 in the SAME-padding halo (branchless)
            const float mask = (hok & (wx >= 0) & (wx < W)) ? 1.0f : 0.0f;
            const float* ip = in + (size_t)((b * H + hyc) * W + wxc) * C;
            const float* wp = wgt + ((dy + 1) * 3 + (dx + 1)) * C * C;
            #pragma unroll
            for (int kc = 0; kc < C; kc += 4) {
                const int ka = kc + hlf * 2;           // this lane's K pair
                const float t0 = ip[ka];               // unconditional loads
                const float t1 = ip[ka + 1];
                v2f a;
                a.x = t0 * mask;
                a.y = t1 * mask;
                v2f b0, b1;
                b0.x = wp[ ka      * C + l16];
                b0.y = wp[(ka + 1) * C + l16];
                b1.x = wp[ ka      * C + 16 + l16];
                b1.y = wp[(ka + 1) * C + 16 + l16];
                acc0 = __builtin_amdgcn_wmma_f32_16x16x4_f32(
                    false, a, false, b0, (short)0, acc0, false, false);
                acc1 = __builtin_amdgcn_wmma_f32_16x16x4_f32(
                    false, a, false, b1, (short)0, acc1, false, false);
            }
        }
    }

    // Epilogue: bias + LeakyReLU(0.3), scatter C/D layout back to NHWC.
    const float bb0 = bias[l16];
    const float bb1 = bias[16 + l16];
    const int pbase = (b * H + h) * W + w0;
    #pragma unroll
    for (int r = 0; r < 8; ++r) {
        const int mm = r + hlf * 8;                    // C/D row for this lane
        float v0 = acc0[r] + bb0;
        v0 = (v0 > 0.0f) ? v0 : LEAKY_ALPHA * v0;
        float v1 = acc1[r] + bb1;
        v1 = (v1 > 0.0f) ? v1 : LEAKY_ALPHA * v1;
        float* op = out + (size_t)(pbase + mm) * C;
        op[l16]      = v0;
        op[16 + l16] = v1;
    }
}

// ---------------------------------------------------------------------------
// FiLM: out[b,h,w,band,c] = x * gamma[b,h,w,c] + beta[b,h,w,c]
// Pure HBM-streaming kernel: 128-bit non-temporal x/out traffic (512 MB total,
// ~22us roofline @ 23.3 TB/s); gamma/beta (4 MB) stay temporal/L2-resident.
// 4096 floats per pixel -> 1024 float4; channel float4 index = v & 7.
// ---------------------------------------------------------------------------
__global__ __launch_bounds__(256) void film_kernel(
    const v4f* __restrict__ x, const v4f* __restrict__ gamma,
    const v4f* __restrict__ beta, v4f* __restrict__ out, int nvec)
{
    const int stride = gridDim.x * blockDim.x;
    for (int v = blockIdx.x * blockDim.x + threadIdx.x; v < nvec; v += stride) {
        const v4f xv = __builtin_nontemporal_load(&x[v]);
        const int g = ((v >> 10) << 3) | (v & 7);
        const v4f gv = gamma[g];
        const v4f bv = beta[g];
        v4f o;
        o.x = fmaf(xv.x, gv.x, bv.x);
        o.y = fmaf(xv.y, gv.y, bv.y);
        o.z = fmaf(xv.z, gv.z, bv.z);
        o.w = fmaf(xv.w, gv.w, bv.w);
        __builtin_nontemporal_store(o, &out[v]);
    }
}

extern "C" void kernel_launch(void* const* d_in, const int* in_sizes, int n_in,
                              void* d_out, int out_size, void* d_ws, size_t ws_size,
                              hipStream_t stream) {
    const float* x   = (const float*)d_in[0];
    const float* psi = (const float*)d_in[1];
    const float* gw1 = (const float*)d_in[2];
    const float* gb1 = (const float*)d_in[3];
    const float* gw2 = (const float*)d_in[4];
    const float* gb2 = (const float*)d_in[5];
    const float* bw1 = (const float*)d_in[6];
    const float* bb1 = (const float*)d_in[7];
    const float* bw2 = (const float*)d_in[8];
    const float* bb2 = (const float*)d_in[9];
    float* out = (float*)d_out;

    const int PIX = 4 * 64 * 64;       // B*H*W = 16384
    const int CH  = 32;

    // Workspace layout: t1g | t1b | gamma | beta, 2 MB each (8 MB total).
    float* t1g   = (float*)d_ws;
    float* t1b   = t1g   + (size_t)PIX * CH;
    float* gam   = t1b   + (size_t)PIX * CH;
    float* bet   = gam   + (size_t)PIX * CH;

    dim3 cgrid(PIX / 16, 2);           // 1024 tiles x 2 branches, 1 wave each
    conv3x3_lrelu_wmma<<<cgrid, 32, 0, stream>>>(psi, psi, gw1, gb1, bw1, bb1, t1g, t1b);
    conv3x3_lrelu_wmma<<<cgrid, 32, 0, stream>>>(t1g, t1b, gw2, gb2, bw2, bb2, gam, bet);

    const int nvec = out_size / 4;     // 16,777,216 float4
    film_kernel<<<16384, 256, 0, stream>>>(
        (const v4f*)x, (const v4f*)gam, (const v4f*)bet, (v4f*)out, nvec);
}